// PointPillarsVoxelization_47064251629647
// MI455X (gfx1250) — compile-verified
//
#include <hip/hip_runtime.h>
#include <hip/hip_bf16.h>

namespace {
constexpr int kNX = 432, kNY = 496, kNZ = 1;
constexpr int kNVOX = kNX * kNY * kNZ;          // 214272 dense voxel cells
constexpr int kMaxPts = 32;
constexpr int kMaxVox = 40000;
constexpr int kCols = 8;                        // 3 xyz + 5 feats
constexpr int kOutVoxElems   = kMaxVox * kMaxPts * kCols;   // 10,240,000
constexpr int kOutCoordElems = kMaxVox * 3;                 // 120,000
constexpr int kOutNumElems   = kMaxVox;                     // 40,000
}

// ---- helpers -------------------------------------------------------------

__device__ __forceinline__ int compute_vid(const float* __restrict__ pts, int i) {
  float x = pts[(size_t)i * kCols + 0];
  float y = pts[(size_t)i * kCols + 1];
  float z = pts[(size_t)i * kCols + 2];
  // Must match JAX: floor((xyz - lo) / vs) in f32 with true division.
  int cx = (int)floorf((x - 0.0f)      / 0.16f);
  int cy = (int)floorf((y - (-39.68f)) / 0.16f);
  int cz = (int)floorf((z - (-3.0f))   / 4.0f);
  if (cx < 0 || cx >= kNX || cy < 0 || cy >= kNY || cz < 0 || cz >= kNZ) return -1;
  return (cz * kNY + cy) * kNX + cx;
}

// ---- kernels -------------------------------------------------------------

__global__ void zero_f4(float4* __restrict__ p, int n4) {
  int i = blockIdx.x * blockDim.x + threadIdx.x;
  int stride = gridDim.x * blockDim.x;
  float4 z; z.x = 0.f; z.y = 0.f; z.z = 0.f; z.w = 0.f;
  for (; i < n4; i += stride) p[i] = z;        // global_store_b128
}

__global__ void zero_i(int* __restrict__ p, int n) {
  int i = blockIdx.x * blockDim.x + threadIdx.x;
  int stride = gridDim.x * blockDim.x;
  for (; i < n; i += stride) p[i] = 0;
}

// Pass 1: dense per-voxel point counts.
__global__ void vox_count(const float* __restrict__ pts, int* __restrict__ cnt, int n) {
  int i = blockIdx.x * blockDim.x + threadIdx.x;
  if (i >= n) return;
  int vid = compute_vid(pts, i);
  if (vid >= 0) atomicAdd(&cnt[vid], 1);
}

// Pass 2: single-workgroup carry-propagating exclusive scan over
// (occupancy flag, count). Wave32-native: 4 items/thread sequential,
// __shfl_up intra-wave scan (no barriers), 32-entry cross-wave LDS scan.
// rankArr = voxel rank in vid order, baseArr = packed point-list offset.
__global__ __launch_bounds__(1024) void vox_scan(const int* __restrict__ cnt,
                                                 int* __restrict__ rankArr,
                                                 int* __restrict__ baseArr) {
  constexpr int ITEMS = 4;
  constexpr int CHUNK = 1024 * ITEMS;           // 4096 -> 53 chunks
  __shared__ int sWF[32], sWC[32];
  __shared__ int sCarry[2];
  __shared__ int sTot[2];
  int tid  = threadIdx.x;
  int lane = tid & 31;
  int wv   = tid >> 5;
  if (tid == 0) { sCarry[0] = 0; sCarry[1] = 0; }
  __syncthreads();
  for (int start = 0; start < kNVOX; start += CHUNK) {
    int cF = sCarry[0], cC = sCarry[1];
    int idx0 = start + tid * ITEMS;
    int cv[ITEMS], fv[ITEMS];
    int sumC = 0, sumF = 0;
#pragma unroll
    for (int k = 0; k < ITEMS; ++k) {
      int i = idx0 + k;
      int c = (i < kNVOX) ? cnt[i] : 0;
      cv[k] = c; fv[k] = (c > 0) ? 1 : 0;
      sumC += c; sumF += fv[k];
    }
    // intra-wave inclusive scan of per-thread sums (wave32 shuffles)
    int incF = sumF, incC = sumC;
#pragma unroll
    for (int off = 1; off < 32; off <<= 1) {
      int nF = __shfl_up(incF, off, 32);
      int nC = __shfl_up(incC, off, 32);
      if (lane >= off) { incF += nF; incC += nC; }
    }
    if (lane == 31) { sWF[wv] = incF; sWC[wv] = incC; }
    __syncthreads();
    // wave 0 scans the 32 wave totals -> exclusive wave offsets + chunk totals
    if (wv == 0) {
      int wF = sWF[lane], wC = sWC[lane];
      int iF = wF, iC = wC;
#pragma unroll
      for (int off = 1; off < 32; off <<= 1) {
        int nF = __shfl_up(iF, off, 32);
        int nC = __shfl_up(iC, off, 32);
        if (lane >= off) { iF += nF; iC += nC; }
      }
      sWF[lane] = iF - wF;
      sWC[lane] = iC - wC;
      if (lane == 31) { sTot[0] = iF; sTot[1] = iC; }
    }
    __syncthreads();
    int baseF = cF + sWF[wv] + (incF - sumF);   // exclusive prefix for item 0
    int baseC = cC + sWC[wv] + (incC - sumC);
#pragma unroll
    for (int k = 0; k < ITEMS; ++k) {
      int i = idx0 + k;
      if (i < kNVOX) { rankArr[i] = baseF; baseArr[i] = baseC; }
      baseF += fv[k]; baseC += cv[k];
    }
    __syncthreads();
    if (tid == 0) { sCarry[0] = cF + sTot[0]; sCarry[1] = cC + sTot[1]; }
    __syncthreads();
  }
}

// Pass 3: scatter point indices into packed per-voxel lists (arrival order
// is nondeterministic; pass 4 re-establishes original-index order exactly).
__global__ void vox_scatter(const float* __restrict__ pts, int* __restrict__ cnt2,
                            const int* __restrict__ baseArr, int* __restrict__ ptidx,
                            int n) {
  int i = blockIdx.x * blockDim.x + threadIdx.x;
  if (i >= n) return;
  int vid = compute_vid(pts, i);
  if (vid >= 0) {
    int p = atomicAdd(&cnt2[vid], 1);
    ptidx[baseArr[vid] + p] = i;
  }
}

// Pass 4: one wave32 per voxel. Fast path (c <= 32): each lane holds one
// point index; stable rank = #{smaller} computed with 32 __shfl broadcasts
// (indices unique => deterministic, matches JAX stable argsort grouping).
// Slow path (c > 32): early-terminating selection over the global list.
__global__ __launch_bounds__(256) void vox_output(
    const float* __restrict__ pts,
    const int* __restrict__ cnt,
    const int* __restrict__ rankArr,
    const int* __restrict__ baseArr,
    const int* __restrict__ ptidx,
    float* __restrict__ out_vox,
    float* __restrict__ out_coords,
    float* __restrict__ out_np) {
  int gid  = blockIdx.x * blockDim.x + threadIdx.x;
  int vid  = gid >> 5;         // wave32: one wave per voxel
  int lane = gid & 31;
  if (vid >= kNVOX) return;
  int c = cnt[vid];
  if (c == 0) return;
  int rank = rankArr[vid];
  if (rank >= kMaxVox) return;
  int base = baseArr[vid];
  int kept = c < kMaxPts ? c : kMaxPts;

  if (lane == 0) {
    out_np[rank] = (float)kept;
    int vz  = vid / (kNX * kNY);
    int rem = vid - vz * (kNX * kNY);
    int vy  = rem / kNX;
    int vx  = rem - vy * kNX;
    out_coords[rank * 3 + 0] = (float)vz;
    out_coords[rank * 3 + 1] = (float)vy;
    out_coords[rank * 3 + 2] = (float)vx;
  }

  if (c <= kMaxPts) {
    // register-resident stable ranking across the wave
    int v = (lane < c) ? ptidx[base + lane] : 0x7fffffff;
    int r = 0;
#pragma unroll
    for (int j = 0; j < 32; ++j) {
      int vj = __shfl(v, j, 32);
      r += (vj < v) ? 1 : 0;
    }
    if (lane < c) {
      const float* src = pts + (size_t)v * kCols;
      float4 a = ((const float4*)src)[0];
      float4 b = ((const float4*)src)[1];
      float* dst = out_vox + ((size_t)rank * kMaxPts + r) * kCols;
      ((float4*)dst)[0] = a;
      ((float4*)dst)[1] = b;
    }
  } else {
    for (int j = lane; j < c; j += 32) {
      int e = ptidx[base + j];
      const float* src = pts + (size_t)e * kCols;
      __builtin_prefetch(src, 0, 0);           // global_prefetch_b8
      int r = 0;
      for (int k = 0; k < c && r < kMaxPts; ++k)
        r += (ptidx[base + k] < e) ? 1 : 0;
      if (r < kMaxPts) {
        float4 a = ((const float4*)src)[0];
        float4 b = ((const float4*)src)[1];
        float* dst = out_vox + ((size_t)rank * kMaxPts + r) * kCols;
        ((float4*)dst)[0] = a;
        ((float4*)dst)[1] = b;
      }
    }
  }
}

// ---- launcher ------------------------------------------------------------

extern "C" void kernel_launch(void* const* d_in, const int* in_sizes, int n_in,
                              void* d_out, int out_size, void* d_ws, size_t ws_size,
                              hipStream_t stream) {
  const float* pts = (const float*)d_in[0];
  int n = in_sizes[0] / kCols;                 // 1,200,000

  float* out        = (float*)d_out;
  float* out_vox    = out;
  float* out_coords = out + kOutVoxElems;
  float* out_np     = out + kOutVoxElems + kOutCoordElems;

  int* ws      = (int*)d_ws;
  int* cnt     = ws;                 // [kNVOX]
  int* cnt2    = ws + kNVOX;         // [kNVOX]
  int* rankArr = ws + 2 * kNVOX;     // [kNVOX]
  int* baseArr = ws + 3 * kNVOX;     // [kNVOX]
  int* ptidx   = ws + 4 * kNVOX;     // [n]  (~8.2 MB total workspace)

  // Re-zero everything each call (d_out/d_ws are poisoned; graph-replay safe).
  int totalOut = kOutVoxElems + kOutCoordElems + kOutNumElems;   // 10,400,000 (mult of 4)
  zero_f4<<<1024, 256, 0, stream>>>((float4*)out, totalOut / 4);
  zero_i <<<512, 256, 0, stream>>>(cnt, 2 * kNVOX);

  const int tpb = 256;
  vox_count  <<<(n + tpb - 1) / tpb, tpb, 0, stream>>>(pts, cnt, n);
  vox_scan   <<<1, 1024, 0, stream>>>(cnt, rankArr, baseArr);
  vox_scatter<<<(n + tpb - 1) / tpb, tpb, 0, stream>>>(pts, cnt2, baseArr, ptidx, n);

  long long totalThreads = (long long)kNVOX * 32;
  int blocks = (int)((totalThreads + tpb - 1) / tpb);
  vox_output <<<blocks, tpb, 0, stream>>>(pts, cnt, rankArr, baseArr, ptidx,
                                          out_vox, out_coords, out_np);
}